// CBOWModel_90263032693057
// MI455X (gfx1250) — compile-verified
//
#include <hip/hip_runtime.h>

typedef float v2f __attribute__((ext_vector_type(2)));
typedef float v4f __attribute__((ext_vector_type(4)));
typedef float v8f __attribute__((ext_vector_type(8)));

#define DDIM 128
#define CWIN 10
#define KNEG 10
#define BROWS 16384
#define EPSF 1e-10f
#define ROWS_PER_WAVE 16
#define WAVES_PER_BLOCK 2
#define TILE_STRIDE 132                    // 128 + 4 pad -> conflict-free ds_load_b64
#define NGROUPS (BROWS / ROWS_PER_WAVE)    // 1024

__global__ __launch_bounds__(WAVES_PER_BLOCK * 32)
void cbow_loss_kernel(const int* __restrict__ ctx_idx,
                      const int* __restrict__ tgt_idx,
                      const int* __restrict__ neg_idx,
                      const float* __restrict__ in_embed,
                      const float* __restrict__ out_embed,
                      float* __restrict__ partials)
{
    __shared__ float smem[WAVES_PER_BLOCK * 2 * ROWS_PER_WAVE * TILE_STRIDE];
    const int lane = threadIdx.x & 31;
    const int wid  = threadIdx.x >> 5;
    float* ctxT = &smem[wid * 2 * ROWS_PER_WAVE * TILE_STRIDE];
    float* embT = ctxT + ROWS_PER_WAVE * TILE_STRIDE;

    const int group = blockIdx.x * WAVES_PER_BLOCK + wid;
    const int b0 = group * ROWS_PER_WAVE;

    // ---- Phase 1: gather + mean-pool context embeddings -> ctxT[16][128] ----
    // Each lane owns 4 consecutive floats of D: one wave load = one full 512B row.
    // Unroll x2 so ~20 global_load_b128 (10KB) are in flight per wave.
    #pragma unroll 2
    for (int i = 0; i < ROWS_PER_WAVE; ++i) {
        const int b = b0 + i;
        const int* cr = ctx_idx + (size_t)b * CWIN;
        v4f acc = {0.f, 0.f, 0.f, 0.f};
        #pragma unroll
        for (int c = 0; c < CWIN; ++c) {
            size_t row = (size_t)cr[c];
            acc += *(const v4f*)(in_embed + row * DDIM + (size_t)lane * 4);
        }
        acc *= (1.0f / (float)CWIN);
        *(v4f*)(ctxT + i * TILE_STRIDE + lane * 4) = acc;
    }
    __syncthreads();

    // Lane roles for WMMA operands (16x4 A / 4x16 B f32 layouts) and diagonal.
    const int m    = lane & 15;        // A row / B column owned by this lane
    const int koff = (lane >> 4) * 2;  // K sub-offset inside each 4-wide chunk
    const int rsel = (lane < 8) ? lane : (lane - 24);
    const bool diagLane = (lane < 8) || (lane >= 24);

    float myLoss = 0.0f;

    for (int s = 0; s <= KNEG; ++s) {
        // Stage the 16 output-embedding rows for this score column.
        // Load ALL rows into registers first (16 b128 loads outstanding),
        // then spill to LDS — maximizes memory-level parallelism.
        v4f stage[ROWS_PER_WAVE];
        #pragma unroll
        for (int i = 0; i < ROWS_PER_WAVE; ++i) {
            const int b = b0 + i;
            size_t row = (s == 0) ? (size_t)tgt_idx[b]
                                  : (size_t)neg_idx[(size_t)b * KNEG + (s - 1)];
            stage[i] = *(const v4f*)(out_embed + row * DDIM + (size_t)lane * 4);
        }
        #pragma unroll
        for (int i = 0; i < ROWS_PER_WAVE; ++i)
            *(v4f*)(embT + i * TILE_STRIDE + lane * 4) = stage[i];
        __syncthreads();

        // D = sum_k A(16x4) x B(4x16): diagonal gives 16 per-row dot products.
        v8f acc = {0.f, 0.f, 0.f, 0.f, 0.f, 0.f, 0.f, 0.f};
        #pragma unroll
        for (int kb = 0; kb < DDIM / 4; ++kb) {
            const int k = kb * 4 + koff;
            v2f a  = *(const v2f*)(ctxT + m * TILE_STRIDE + k);
            v2f bb = *(const v2f*)(embT + m * TILE_STRIDE + k);
            acc = __builtin_amdgcn_wmma_f32_16x16x4_f32(
                false, a, false, bb, (short)0, acc, false, false);
        }

        // Extract D[i][i]: lanes 0-7 hold rows 0-7 (vgpr=lane),
        // lanes 24-31 hold rows 8-15 (vgpr=lane-24).
        float score = acc[0];
        #pragma unroll
        for (int r = 1; r < 8; ++r) score = (rsel == r) ? acc[r] : score;

        // Branch-free loss so EXEC stays all-ones around the WMMA region.
        float sig = 1.0f / (1.0f + expf(-score));
        float contrib = (s == 0) ? logf(sig + EPSF) : logf(1.0f - sig + EPSF);
        myLoss += diagLane ? contrib : 0.0f;

        __syncthreads();
    }

    // wave32 reduction of the 16 valid lane losses (others contribute 0)
    #pragma unroll
    for (int off = 16; off >= 1; off >>= 1)
        myLoss += __shfl_xor(myLoss, off, 32);
    if (lane == 0) partials[group] = myLoss;
}

__global__ __launch_bounds__(256)
void cbow_reduce_kernel(const float* __restrict__ partials, float* __restrict__ out)
{
    __shared__ float sh[256];
    const int t = threadIdx.x;
    float s = 0.0f;
    for (int i = t; i < NGROUPS; i += 256) s += partials[i];   // fixed order
    sh[t] = s;
    __syncthreads();
    for (int stride = 128; stride >= 1; stride >>= 1) {
        if (t < stride) sh[t] += sh[t + stride];
        __syncthreads();
    }
    if (t == 0) out[0] = -sh[0] / (float)BROWS;
}

extern "C" void kernel_launch(void* const* d_in, const int* in_sizes, int n_in,
                              void* d_out, int out_size, void* d_ws, size_t ws_size,
                              hipStream_t stream) {
    (void)in_sizes; (void)n_in; (void)out_size; (void)ws_size;
    const int*   ctx = (const int*)d_in[0];    // [B, C]
    const int*   tgt = (const int*)d_in[1];    // [B]
    const int*   neg = (const int*)d_in[2];    // [B, K]
    const float* ine = (const float*)d_in[3];  // [V, D]
    const float* oue = (const float*)d_in[4];  // [V, D]
    float* partials = (float*)d_ws;            // NGROUPS floats

    dim3 grid(NGROUPS / WAVES_PER_BLOCK);
    dim3 block(WAVES_PER_BLOCK * 32);
    cbow_loss_kernel<<<grid, block, 0, stream>>>(ctx, tgt, neg, ine, oue, partials);
    cbow_reduce_kernel<<<1, 256, 0, stream>>>(partials, (float*)d_out);
}